// MyModel_56573309223171
// MI455X (gfx1250) — compile-verified
//
#include <hip/hip_runtime.h>

// Problem constants (reference: T=2048, V=50000, D=H=25)
#define T_  2048
#define V_  50000
#define D_  25
#define H_  25
#define NG  100      // 4*H gates
#define BK  128      // K-chunk staged in LDS
#define BKP 132      // LDS row stride (floats), padded for bank-conflict-free b64 reads
#define KSPLIT 20
#define KSLICE 2500  // 2500 = 19*128 + 68 ; tail (68) is a multiple of 4 -> clean WMMA K-steps
#define NCHUNK 20    // chunks per slice (19 full + 1 tail)
#define KTAIL  68

typedef __attribute__((ext_vector_type(2))) float v2f;
typedef __attribute__((ext_vector_type(8))) float v8f;

// ---------------------------------------------------------------------------
// Kernel 1: partial[slice][T][32] = inputs[:, kslice] @ emb[kslice, :]  (fp32 WMMA)
// Block = 256 threads (8 waves). Wave w owns M-tile rows blockIdx.x*128 + w*16.
// blockIdx.y = K-slice. B double-buffered in LDS as [2][n=32][BKP], cols >=25 zero.
// ---------------------------------------------------------------------------
__device__ __forceinline__ void kchunk(const float* __restrict__ inputs,
                                       const float* __restrict__ ldsB,
                                       size_t aBase, int kb, int kc,
                                       int bOff0, int bOff1,
                                       v8f& acc0, v8f& acc1) {
#pragma unroll 8
  for (int kk = 0; kk < kc; kk += 4) {
    if ((kk & 31) == 0) {
      // Prefetch the A stream one chunk (512B/row-pair) ahead; clamp to buffer.
      size_t pidx = aBase + (size_t)(kb + kk + BK);
      const size_t last = (size_t)T_ * V_ - 2;
      if (pidx > last) pidx = last;
      __builtin_prefetch(inputs + pidx, 0, 1);   // -> global_prefetch_b8
    }
    // A frag: lane (0..15) = row M, lane>=16 holds K+2 ; v0=K, v1=K+1
    v2f a  = *(const v2f*)(inputs + aBase + kb + kk);
    // B frags (from LDS, transposed layout): v0 = B[K][n], v1 = B[K+1][n]
    v2f b0 = *(const v2f*)(ldsB + bOff0 + kk);
    v2f b1 = *(const v2f*)(ldsB + bOff1 + kk);
    acc0 = __builtin_amdgcn_wmma_f32_16x16x4_f32(false, a, false, b0,
                                                 (short)0, acc0, false, false);
    acc1 = __builtin_amdgcn_wmma_f32_16x16x4_f32(false, a, false, b1,
                                                 (short)0, acc1, false, false);
  }
}

// Staging split into load- and store-phases so the global loads for chunk ch+1
// stay in flight underneath chunk ch's 64 WMMAs (software pipelining).
__device__ __forceinline__ void stage_load(const float* __restrict__ emb,
                                           int kb, int kEnd, int tn, int tk0,
                                           float v[16]) {
#pragma unroll
  for (int i = 0; i < 16; ++i) {
    const int gk = kb + tk0 + i * 8;
    v[i] = (tn < 25 && gk < kEnd) ? emb[(size_t)gk * 25 + tn] : 0.f;
  }
}

__device__ __forceinline__ void stage_store(float* __restrict__ ldsB,
                                            int tn, int tk0, const float v[16]) {
#pragma unroll
  for (int i = 0; i < 16; ++i)
    ldsB[tn * BKP + tk0 + i * 8] = v[i];
}

__global__ __launch_bounds__(256) void emb_gemm_part(
    const float* __restrict__ inputs,   // [T, V]
    const float* __restrict__ emb,      // [V, 25]
    float* __restrict__ part)           // [KSPLIT, T, 32]
{
  __shared__ float ldsB[2][32 * BKP];

  const int tid  = threadIdx.x;
  const int lane = tid & 31;
  const int wave = tid >> 5;
  const int hi   = lane >> 4;     // selects K+2 half
  const int nn   = lane & 15;     // M for A-frag, tile-local N for B/C frags

  const int mBase = blockIdx.x * 128 + wave * 16;
  const size_t aBase = (size_t)(mBase + nn) * V_ + 2 * hi;
  const int bOff0 = nn * BKP + 2 * hi;
  const int bOff1 = (16 + nn) * BKP + 2 * hi;

  const int kbS  = blockIdx.y * KSLICE;
  const int kEnd = kbS + KSLICE;

  // staging coords: tn = column n (0..31), rows strided by 8
  const int tn  = tid & 31;
  const int tk0 = tid >> 5;

  v8f acc0 = {0.f,0.f,0.f,0.f,0.f,0.f,0.f,0.f};
  v8f acc1 = {0.f,0.f,0.f,0.f,0.f,0.f,0.f,0.f};

  float sv[16];
  stage_load(emb, kbS, kEnd, tn, tk0, sv);     // prologue: chunk 0
  stage_store(&ldsB[0][0], tn, tk0, sv);

  for (int ch = 0; ch < NCHUNK; ++ch) {
    __syncthreads();                            // buf[ch&1] staged; buf[(ch+1)&1] free
    const float* cur = &ldsB[ch & 1][0];
    float*       nxt = &ldsB[(ch + 1) & 1][0];
    const int kb = kbS + ch * BK;
    if (ch + 1 < NCHUNK)
      stage_load(emb, kb + BK, kEnd, tn, tk0, sv);   // loads in flight over WMMAs
    if (ch < NCHUNK - 1)
      kchunk(inputs, cur, aBase, kb, BK,    bOff0, bOff1, acc0, acc1);
    else
      kchunk(inputs, cur, aBase, kb, KTAIL, bOff0, bOff1, acc0, acc1);
    if (ch + 1 < NCHUNK)
      stage_store(nxt, tn, tk0, sv);
  }

  // C layout: VGPR r -> lanes 0-15: (M=r, N=lane) ; lanes 16-31: (M=8+r, N=lane-16)
  float* p = part + (size_t)blockIdx.y * T_ * 32;
#pragma unroll
  for (int r = 0; r < 8; ++r) {
    const int row = mBase + r + 8 * hi;
    p[row * 32 + nn]      = acc0[r];
    p[row * 32 + 16 + nn] = acc1[r];           // cols 25..31 are exact zeros
  }
}

// ---------------------------------------------------------------------------
// Kernel 2: embsP[T][32] = sum over K-slices (deterministic fixed-order reduce)
// ---------------------------------------------------------------------------
__global__ __launch_bounds__(256) void reduce_embs(const float* __restrict__ part,
                                                   float* __restrict__ embsP) {
  const int id = blockIdx.x * 256 + threadIdx.x;   // T_*32 = 65536 ids
  float s = 0.f;
#pragma unroll
  for (int k = 0; k < KSPLIT; ++k) s += part[(size_t)k * T_ * 32 + id];
  embsP[id] = s;
}

// ---------------------------------------------------------------------------
// Kernel 3: Gx[t][j] = b_ih[j] + b_hh[j] + embs[t,:] . W_ih[j,:]   (tiny GEMM)
// ---------------------------------------------------------------------------
__global__ __launch_bounds__(256) void gx_kernel(const float* __restrict__ embsP,
                                                 const float* __restrict__ W_ih,
                                                 const float* __restrict__ b_ih,
                                                 const float* __restrict__ b_hh,
                                                 float* __restrict__ Gx) {
  const int id = blockIdx.x * 256 + threadIdx.x;   // T_*NG = 204800 ids
  const int t = id / NG, j = id % NG;
  float acc = b_ih[j] + b_hh[j];
#pragma unroll
  for (int d = 0; d < D_; ++d)
    acc += embsP[t * 32 + d] * W_ih[j * D_ + d];
  Gx[id] = acc;
}

// ---------------------------------------------------------------------------
// Kernel 4: sequential LSTM scan. One workgroup (4 waves on one WGP).
// Thread j (<100) owns gate j; W_hh row resident in registers; h & gates in LDS.
// ---------------------------------------------------------------------------
__global__ __launch_bounds__(128) void lstm_scan(const float* __restrict__ Gx,
                                                 const float* __restrict__ W_hh,
                                                 float* __restrict__ out) {
  __shared__ float hbuf[32];
  __shared__ float gact[NG];
  const int j = threadIdx.x;

  float w[D_];
  if (j < NG) {
#pragma unroll
    for (int d = 0; d < D_; ++d) w[d] = W_hh[j * D_ + d];
  }
  float c = 0.f;
  if (j < 32) hbuf[j] = 0.f;
  __syncthreads();

  for (int t = 0; t < T_; ++t) {
    if (j < NG) {
      float acc = Gx[t * NG + j];
#pragma unroll
      for (int d = 0; d < D_; ++d) acc += w[d] * hbuf[d];   // LDS broadcast reads
      float a;
      if (j < 50 || j >= 75) a = 1.f / (1.f + __expf(-acc)); // i, f, o : sigmoid
      else                   a = tanhf(acc);                 // g : tanh
      gact[j] = a;
    }
    __syncthreads();
    if (j < H_) {
      const float ig = gact[j], fg = gact[25 + j], gg = gact[50 + j], og = gact[75 + j];
      c = fg * c + ig * gg;
      const float h = og * tanhf(c);
      hbuf[j] = h;
      out[t * H_ + j] = h;
    }
    __syncthreads();
  }
}

// ---------------------------------------------------------------------------
extern "C" void kernel_launch(void* const* d_in, const int* in_sizes, int n_in,
                              void* d_out, int out_size, void* d_ws, size_t ws_size,
                              hipStream_t stream) {
  const float* inputs = (const float*)d_in[0];   // [T, V]
  const float* emb    = (const float*)d_in[1];   // [V, 25]
  const float* W_ih   = (const float*)d_in[2];   // [100, 25]
  const float* W_hh   = (const float*)d_in[3];   // [100, 25]
  const float* b_ih   = (const float*)d_in[4];   // [100]
  const float* b_hh   = (const float*)d_in[5];   // [100]
  float* out = (float*)d_out;                    // [T, 25]

  // Workspace layout (~6.3 MB): partials | embsP | Gx
  float* part  = (float*)d_ws;                   // KSPLIT * T_ * 32
  float* embsP = part + (size_t)KSPLIT * T_ * 32;// T_ * 32
  float* Gx    = embsP + (size_t)T_ * 32;        // T_ * NG

  dim3 g1(T_ / 128, KSPLIT);                     // 16 x 20 blocks, 8 waves each
  emb_gemm_part<<<g1, 256, 0, stream>>>(inputs, emb, part);
  reduce_embs<<<(T_ * 32) / 256, 256, 0, stream>>>(part, embsP);
  gx_kernel<<<(T_ * NG) / 256, 256, 0, stream>>>(embsP, W_ih, b_ih, b_hh, Gx);
  lstm_scan<<<1, 128, 0, stream>>>(Gx, W_hh, out);
}